// RivTreeCluster_48352741818707
// MI455X (gfx1250) — compile-verified
//
#include <hip/hip_runtime.h>

// RivTreeCluster: out = x; out[dst_idx[g]] += x[src_idx[g]] for g in [0,E)
// Pure bandwidth problem (~1.3 GB traffic, ~0.025 FLOP/byte) -> no WMMA path
// applies. Optimal MI455X kernel = b128 vectorized streaming copy (NT hints)
// + wave32-per-transfer coalesced gather + hardware f32 atomics at L2.

#define ROW_T 128  // floats per node row (T in the reference)

// Native clang vector type: legal operand for __builtin_nontemporal_* and
// lowers to b128 global memory ops (HIP's float4 class is rejected).
typedef __attribute__((ext_vector_type(4))) float v4f;

// ---------------------------------------------------------------------------
// Phase 1: out = x. One 16B vector per thread, non-temporal load+store
// (512 MB stream, 2.7x the 192 MB L2 -> no caching benefit).
// ---------------------------------------------------------------------------
__global__ __launch_bounds__(256) void rivtree_copy_kernel(
    const v4f* __restrict__ x4, v4f* __restrict__ out4, long long n4) {
  long long i = (long long)blockIdx.x * blockDim.x + threadIdx.x;
  if (i < n4) {
    v4f v = __builtin_nontemporal_load(&x4[i]);
    __builtin_nontemporal_store(v, &out4[i]);
  }
}

// ---------------------------------------------------------------------------
// Phase 2: out[dst[g]*128 .. +128) += x[src[g]*128 .. +128).
// One wave32 per transfer g: lane L handles floats [4L, 4L+4) of the row.
//   - gather: global_load_b128, wave reads one contiguous 512B row (coalesced)
//   - scatter: 4x global_atomic_add_f32 (non-returning, L2-side RMW);
//     atomics required because dst_idx has duplicates that must accumulate.
// Gather reads the immutable input x, so it never races the atomics on out.
// ---------------------------------------------------------------------------
__global__ __launch_bounds__(256) void rivtree_scatter_kernel(
    const float* __restrict__ x, const int* __restrict__ src_idx,
    const int* __restrict__ dst_idx, float* __restrict__ out, int E) {
  int tid  = blockIdx.x * blockDim.x + threadIdx.x;
  int g    = tid >> 5;          // transfer id (one wave32 per transfer)
  int lane = tid & 31;          // lane within wave
  if (g >= E) return;

  long long s = (long long)src_idx[g] * ROW_T;
  long long d = (long long)dst_idx[g] * ROW_T;

  // Coalesced 16B gather from the source row (512B per wave, row-aligned).
  const v4f* sp = (const v4f*)(x + s) + lane;
  v4f v = *sp;

  // Hardware f32 atomic adds into the destination row (no return value ->
  // non-returning GLOBAL_ATOMIC_ADD_F32, tracked on STOREcnt).
  float* dp = out + d + (long long)lane * 4;
  unsafeAtomicAdd(dp + 0, v.x);
  unsafeAtomicAdd(dp + 1, v.y);
  unsafeAtomicAdd(dp + 2, v.z);
  unsafeAtomicAdd(dp + 3, v.w);
}

extern "C" void kernel_launch(void* const* d_in, const int* in_sizes, int n_in,
                              void* d_out, int out_size, void* d_ws, size_t ws_size,
                              hipStream_t stream) {
  (void)n_in; (void)d_ws; (void)ws_size;

  const float* x       = (const float*)d_in[0];   // [TOTAL_NODES * 128] f32
  const int*   src_idx = (const int*)d_in[1];     // [E] indices
  const int*   dst_idx = (const int*)d_in[2];     // [E] indices
  float*       out     = (float*)d_out;           // [TOTAL_NODES * 128] f32

  long long n_elem = (long long)out_size;         // TOTAL_NODES * 128
  long long n4     = n_elem >> 2;                 // 16B-vector count
  int       E      = in_sizes[1];                 // transfer count

  // Phase 1: streaming copy out = x (fully overwrites poisoned d_out).
  {
    long long blocks = (n4 + 255) / 256;
    rivtree_copy_kernel<<<dim3((unsigned)blocks), dim3(256), 0, stream>>>(
        (const v4f*)x, (v4f*)out, n4);
  }

  // Phase 2: scatter-add (stream-ordered after the copy completes).
  {
    long long threads = (long long)E * 32;        // one wave32 per transfer
    long long blocks  = (threads + 255) / 256;
    rivtree_scatter_kernel<<<dim3((unsigned)blocks), dim3(256), 0, stream>>>(
        x, src_idx, dst_idx, out, E);
  }
}